// NeighborhoodSelfAttentionBlock_7481833029937
// MI455X (gfx1250) — compile-verified
//
#include <hip/hip_runtime.h>
#include <math.h>

typedef float v2f __attribute__((ext_vector_type(2)));
typedef float v8f __attribute__((ext_vector_type(8)));

#define TOKENS 4096   // N*H*W = 4*32*32
#define C_DIM  512
#define EPSF   1e-6f

#define WMMA_F32(a, b, c) \
  __builtin_amdgcn_wmma_f32_16x16x4_f32(false, (a), false, (b), (short)0, (c), false, false)

// CDNA5 async global->LDS copy (ASYNCcnt). lds_off = byte offset in LDS
// (low 32 bits of the generic pointer == LDS offset per the aperture rules).
__device__ __forceinline__ void async_copy_b128(unsigned lds_off, const float* g) {
  asm volatile("global_load_async_to_lds_b128 %0, %1, off"
               :: "v"(lds_off), "v"(g) : "memory");
}
__device__ __forceinline__ unsigned lds_off32(const void* p) {
  return (unsigned)(unsigned long long)p;
}

// ---------------------------------------------------------------------------
// Kernel 0: s[c] = 1 + sum_k cond[k] * norm_w[k][c]   (cond repeated over T)
// ---------------------------------------------------------------------------
__global__ void k_cond_scale(const float* __restrict__ cond,
                             const float* __restrict__ norm_w,
                             float* __restrict__ s_vec) {
  int t = threadIdx.x;
  for (int c = t; c < C_DIM; c += 256) {
    float acc = 1.0f;
    for (int k = 0; k < 256; ++k) acc += cond[k] * norm_w[k * C_DIM + c];
    s_vec[c] = acc;
  }
}

// ---------------------------------------------------------------------------
// Kernel 1: RMS-norm + scale with LDS-tiled transpose x(C,THW) -> xn(token,C).
// Block = 256 threads, 32 tokens; coalesced reads of x, coalesced writes of xn.
// ---------------------------------------------------------------------------
__global__ void k_rmsnorm(const float* __restrict__ x,
                          const float* __restrict__ s_vec,
                          float* __restrict__ xn) {
  __shared__ float tile[32][513];
  __shared__ float psum[8][32];
  __shared__ float rstdS[32];
  int t = threadIdx.x;
  int tok0 = blockIdx.x * 32;
  int tok = t & 31;
  int crow = t >> 5;  // 0..7
  for (int pass = 0; pass < 64; ++pass) {
    int c = pass * 8 + crow;
    tile[tok][c] = x[(size_t)c * TOKENS + tok0 + tok];   // coalesced over tok
  }
  __syncthreads();
  float ps = 0.f;
  int cbase = crow * 64;
  for (int j = 0; j < 64; ++j) { float v = tile[tok][cbase + j]; ps += v * v; }
  psum[crow][tok] = ps;
  __syncthreads();
  if (t < 32) {
    float tot = 0.f;
#pragma unroll
    for (int p = 0; p < 8; ++p) tot += psum[p][t];
    rstdS[t] = rsqrtf(tot * (1.0f / C_DIM) + EPSF);
  }
  __syncthreads();
  int wtok = t >> 3;       // 0..31
  int clane = t & 7;       // interleaved channels -> conflict-free, coalesced
  float r = rstdS[wtok];
  for (int j = 0; j < 64; ++j) {
    int c = clane + j * 8;
    xn[(size_t)(tok0 + wtok) * C_DIM + c] = tile[wtok][c] * s_vec[c] * r;
  }
}

// ---------------------------------------------------------------------------
// Kernel 2/5: fp32 WMMA GEMM, async double-buffered LDS staging.
// C(MxN) = A(MxK) * B(KxN), row-major. Block 256 thr = 8 waves (4M x 2N),
// wave tile 32x64, block tile 128x128, K chunk = 16.
// ---------------------------------------------------------------------------
#define APAD 20    // A tile row stride (16 data + 4 pad), 16B-multiple
#define BPAD 136   // B tile row stride (128 data + 8 pad), 16B-multiple
__global__ void k_gemm_wmma(const float* __restrict__ A,
                            const float* __restrict__ B,
                            float* __restrict__ C,
                            int K, int N) {
  __shared__ float LA[2][128 * APAD];
  __shared__ float LB[2][16 * BPAD];

  int tid  = threadIdx.x;
  int lane = tid & 31;
  int wv   = tid >> 5;
  int row  = lane & 15;
  int koff = (lane >> 4) << 1;
  int m0 = (wv >> 1) * 32;   // wave M origin inside block tile
  int n0 = (wv & 1) * 64;    // wave N origin inside block tile
  int mb = blockIdx.y * 128;
  int nb = blockIdx.x * 128;

  // per-chunk async prefetch: 4 x b128 per thread (2 for A, 2 for B)
  int qa = tid * 2;
  int arow0 = (qa) >> 2, aseg0 = (qa) & 3;
  int arow1 = (qa + 1) >> 2, aseg1 = (qa + 1) & 3;
  int brow0 = (qa) >> 5, bcol0 = (qa) & 31;
  int brow1 = (qa + 1) >> 5, bcol1 = (qa + 1) & 31;

  v8f acc0[4] = {}, acc1[4] = {};
  int NC = K >> 4;

  // prefetch chunk 0 -> buffer 0
  {
    async_copy_b128(lds_off32(&LA[0][arow0 * APAD + aseg0 * 4]),
                    A + (size_t)(mb + arow0) * K + aseg0 * 4);
    async_copy_b128(lds_off32(&LA[0][arow1 * APAD + aseg1 * 4]),
                    A + (size_t)(mb + arow1) * K + aseg1 * 4);
    async_copy_b128(lds_off32(&LB[0][brow0 * BPAD + bcol0 * 4]),
                    B + (size_t)brow0 * N + nb + bcol0 * 4);
    async_copy_b128(lds_off32(&LB[0][brow1 * BPAD + bcol1 * 4]),
                    B + (size_t)brow1 * N + nb + bcol1 * 4);
  }

  for (int c = 0; c < NC; ++c) {
    if (c + 1 < NC) {
      int k0 = (c + 1) << 4;
      int nbuf = (c + 1) & 1;
      async_copy_b128(lds_off32(&LA[nbuf][arow0 * APAD + aseg0 * 4]),
                      A + (size_t)(mb + arow0) * K + k0 + aseg0 * 4);
      async_copy_b128(lds_off32(&LA[nbuf][arow1 * APAD + aseg1 * 4]),
                      A + (size_t)(mb + arow1) * K + k0 + aseg1 * 4);
      async_copy_b128(lds_off32(&LB[nbuf][brow0 * BPAD + bcol0 * 4]),
                      B + (size_t)(k0 + brow0) * N + nb + bcol0 * 4);
      async_copy_b128(lds_off32(&LB[nbuf][brow1 * BPAD + bcol1 * 4]),
                      B + (size_t)(k0 + brow1) * N + nb + bcol1 * 4);
      asm volatile("s_wait_asynccnt 0x4" ::: "memory");  // chunk c landed
    } else {
      asm volatile("s_wait_asynccnt 0x0" ::: "memory");
    }
    __syncthreads();  // chunk-c data visible block-wide

    const float* la = &LA[c & 1][0];
    const float* lb = &LB[c & 1][0];
#pragma unroll
    for (int ks = 0; ks < 4; ++ks) {
      int kb = ks * 4 + koff;
      v2f a0, a1;
      a0.x = la[(m0 + row) * APAD + kb];
      a0.y = la[(m0 + row) * APAD + kb + 1];
      a1.x = la[(m0 + 16 + row) * APAD + kb];
      a1.y = la[(m0 + 16 + row) * APAD + kb + 1];
      const float* bbp = lb + kb * BPAD + n0 + row;
#pragma unroll
      for (int i = 0; i < 4; ++i) {
        v2f b;
        b.x = bbp[i * 16];
        b.y = bbp[i * 16 + BPAD];
        acc0[i] = WMMA_F32(a0, b, acc0[i]);
        acc1[i] = WMMA_F32(a1, b, acc1[i]);
      }
    }
    __syncthreads();  // all waves done reading before buffer reuse
  }

  int mr = (lane >> 4) << 3;
#pragma unroll
  for (int mt = 0; mt < 2; ++mt) {
#pragma unroll
    for (int i = 0; i < 4; ++i) {
      v8f acc = mt ? acc1[i] : acc0[i];
      int n = nb + n0 + i * 16 + row;
#pragma unroll
      for (int e = 0; e < 8; ++e) {
        int m = mb + m0 + mt * 16 + mr + e;
        C[(size_t)m * N + n] = acc[e];
      }
    }
  }
}

// ---------------------------------------------------------------------------
// Kernel 3: per-(token,head) q/k L2-normalize * sqrt(scale), then RoPE.
// One wave per (token, head). Lane l owns dims l and l+32. In-place on qkv.
// ---------------------------------------------------------------------------
__global__ void k_qknorm_rope(float* __restrict__ qkv,
                              const float* __restrict__ pos,
                              const float* __restrict__ scale) {
  int wv = threadIdx.x >> 5;
  int l  = threadIdx.x & 31;
  int gw = blockIdx.x * 8 + wv;
  int token = gw >> 3;
  int head  = gw & 7;
  int hh = (token >> 5) & 31, ww = token & 31;

  float p0 = pos[(hh * 32 + ww) * 2 + 0];
  float p1 = pos[(hh * 32 + ww) * 2 + 1];
  int i  = l & 15;
  int i8 = i & 7;
  float fr = 3.14159265358979323846f *
             expf((float)(i8 * 8 + head) * (2.302585092994045684f / 64.0f));
  float th = (i < 8 ? p0 : p1) * fr;
  float cs = cosf(th), sn = sinf(th);
  float ssc = sqrtf(scale[head]);

  float* q  = qkv + token * 1536 + head * 64;
  float* kk = q + 512;

#pragma unroll
  for (int which = 0; which < 2; ++which) {
    float* v = which ? kk : q;
    float va = v[l], vb = v[l + 32];
    float ss = va * va + vb * vb;
#pragma unroll
    for (int m = 16; m >= 1; m >>= 1) ss += __shfl_xor(ss, m, 32);
    float r = ssc * rsqrtf(ss + EPSF);
    va *= r; vb *= r;
    float vp = __shfl_xor(va, 16, 32);
    float vn = (l < 16) ? (va * cs - vp * sn) : (va * cs + vp * sn);
    v[l] = vn;
    v[l + 32] = vb;
  }
}

// ---------------------------------------------------------------------------
// Kernel 4: neighborhood attention. One wave per (token, head).
// ---------------------------------------------------------------------------
__global__ void k_attn(const float* __restrict__ qkv, float* __restrict__ O) {
  __shared__ float lq[8][64];
  __shared__ float lat[8][64];
  __shared__ int   lkt[8][64];
  int wv = threadIdx.x >> 5;
  int l  = threadIdx.x & 31;
  int gw = blockIdx.x * 8 + wv;
  int token = gw >> 3;
  int head  = gw & 7;
  int n = token >> 10, hh = (token >> 5) & 31, ww = token & 31;
  int ih0 = min(max(hh - 3, 0), 25);
  int iw0 = min(max(ww - 3, 0), 25);

  const float* qb = qkv + token * 1536 + head * 64;
  lq[wv][l] = qb[l];
  lq[wv][l + 32] = qb[l + 32];
  int j0 = l, j1 = l + 32;
  lkt[wv][j0] = n * 1024 + (ih0 + j0 / 7) * 32 + (iw0 + j0 % 7);
  lkt[wv][j1] = (j1 < 49) ? (n * 1024 + (ih0 + j1 / 7) * 32 + (iw0 + j1 % 7)) : 0;
  __syncthreads();

  const float* kb0 = qkv + lkt[wv][j0] * 1536 + 512 + head * 64;
  const float* kb1 = qkv + lkt[wv][j1] * 1536 + 512 + head * 64;
  float a0 = 0.f, a1 = 0.f;
  for (int d = 0; d < 64; ++d) {
    float qd = lq[wv][d];
    a0 += qd * kb0[d];
    a1 += qd * kb1[d];
  }
  float s0 = a0;
  float s1 = (j1 < 49) ? a1 : -1e30f;

  float mx = fmaxf(s0, s1);
#pragma unroll
  for (int m = 16; m >= 1; m >>= 1) mx = fmaxf(mx, __shfl_xor(mx, m, 32));
  float e0 = expf(s0 - mx);
  float e1 = (j1 < 49) ? expf(s1 - mx) : 0.f;
  float sum = e0 + e1;
#pragma unroll
  for (int m = 16; m >= 1; m >>= 1) sum += __shfl_xor(sum, m, 32);
  float inv = 1.0f / sum;
  lat[wv][j0] = e0 * inv;
  lat[wv][j1] = e1 * inv;
  __syncthreads();

  float o0 = 0.f, o1 = 0.f;
  for (int j = 0; j < 49; ++j) {
    const float* vb = qkv + lkt[wv][j] * 1536 + 1024 + head * 64;
    float aj = lat[wv][j];
    o0 += aj * vb[l];
    o1 += aj * vb[l + 32];
  }
  O[token * C_DIM + head * 64 + l]      = o0;
  O[token * C_DIM + head * 64 + l + 32] = o1;
}

// ---------------------------------------------------------------------------
// Kernel 6: epilogue transpose + skip. Yg(token,C) -> out(C,THW), += x.
// 64x64 tile via LDS; coalesced reads and writes.
// ---------------------------------------------------------------------------
__global__ void k_epilogue(const float* __restrict__ Yg,
                           const float* __restrict__ x,
                           float* __restrict__ out) {
  __shared__ float tile[64][65];
  int t = threadIdx.x;
  int tok0 = blockIdx.x * 64;
  int c0   = blockIdx.y * 64;
  int ltok = t >> 4;
  int lc4  = (t & 15) * 4;
#pragma unroll
  for (int i = 0; i < 4; ++i) {
    int tok = ltok + i * 16;
    const float4 v = *(const float4*)(Yg + (size_t)(tok0 + tok) * C_DIM + c0 + lc4);
    tile[tok][lc4 + 0] = v.x;
    tile[tok][lc4 + 1] = v.y;
    tile[tok][lc4 + 2] = v.z;
    tile[tok][lc4 + 3] = v.w;
  }
  __syncthreads();
  int lc  = t >> 4;
  int lt4 = (t & 15) * 4;
#pragma unroll
  for (int i = 0; i < 4; ++i) {
    int c = lc + i * 16;
    size_t o = (size_t)(c0 + c) * TOKENS + tok0 + lt4;
    float4 xs = *(const float4*)(x + o);
    float4 r;
    r.x = tile[lt4 + 0][c] + xs.x;
    r.y = tile[lt4 + 1][c] + xs.y;
    r.z = tile[lt4 + 2][c] + xs.z;
    r.w = tile[lt4 + 3][c] + xs.w;
    *(float4*)(out + o) = r;
  }
}

// ---------------------------------------------------------------------------
extern "C" void kernel_launch(void* const* d_in, const int* in_sizes, int n_in,
                              void* d_out, int out_size, void* d_ws, size_t ws_size,
                              hipStream_t stream) {
  const float* x      = (const float*)d_in[0];   // (1,512,4,32,32)
  const float* pos    = (const float*)d_in[1];   // (1,32,32,2)
  const float* cond   = (const float*)d_in[2];   // (1,256)
  const float* norm_w = (const float*)d_in[3];   // (256,512)
  const float* qkv_w  = (const float*)d_in[4];   // (512,1536)
  const float* scale  = (const float*)d_in[5];   // (8,)
  const float* out_w  = (const float*)d_in[6];   // (512,512)
  float* out = (float*)d_out;                    // (1,512,4,32,32)
  float* ws  = (float*)d_ws;

  float* s_vec = ws;                                 // 512
  float* xn    = ws + 1024;                          // 4096*512
  float* qkv   = xn + (size_t)TOKENS * C_DIM;        // 4096*1536
  float* O     = xn;                                 // reuse after qkv GEMM
  float* Yg    = qkv;                                // reuse after attention

  k_cond_scale<<<1, 256, 0, stream>>>(cond, norm_w, s_vec);
  k_rmsnorm<<<TOKENS / 32, 256, 0, stream>>>(x, s_vec, xn);
  k_gemm_wmma<<<dim3(1536 / 128, TOKENS / 128), 256, 0, stream>>>(
      xn, qkv_w, qkv, 512, 1536);
  k_qknorm_rope<<<TOKENS, 256, 0, stream>>>(qkv, pos, scale);
  k_attn<<<TOKENS, 256, 0, stream>>>(qkv, O);
  k_gemm_wmma<<<dim3(512 / 128, TOKENS / 128), 256, 0, stream>>>(
      O, out_w, Yg, 512, 512);
  k_epilogue<<<dim3(TOKENS / 64, C_DIM / 64), 256, 0, stream>>>(Yg, x, out);
}